// GIN_63608465654042
// MI455X (gfx1250) — compile-verified
//
#include <hip/hip_runtime.h>

#define NN 50000
#define NE 800000
#define HID 256
#define LDSTR (HID + 4)   // padded LDS row stride (floats) -> conflict-free frag reads

typedef __attribute__((ext_vector_type(2))) float v2f;
typedef __attribute__((ext_vector_type(8))) float v8f;

// ---------------- zero fill (float4) ----------------
__global__ void gin_zero_kernel(float4* __restrict__ p, int n4) {
    int i = blockIdx.x * blockDim.x + threadIdx.x;
    if (i < n4) p[i] = make_float4(0.f, 0.f, 0.f, 0.f);
}

// ---------------- weight pack: Wp[k2][n] = {W[2k2][n], W[2k2+1][n]} ----------------
// HID==256 columns fixed. kHalf = K/2 rows of pairs.
__global__ void gin_pack_kernel(const float* __restrict__ w, float2* __restrict__ wp, int kHalf) {
    int i = blockIdx.x * blockDim.x + threadIdx.x;
    if (i >= kHalf * HID) return;
    int k2 = i >> 8;
    int n  = i & 255;
    wp[i] = make_float2(w[(size_t)(2 * k2) * HID + n], w[(size_t)(2 * k2 + 1) * HID + n]);
}

// ---------------- edge scatter-add: agg[dst] += h[src] ----------------
template <int DIM>
__global__ void gin_scatter_kernel(const float* __restrict__ h,
                                   const int* __restrict__ src,
                                   const int* __restrict__ dst,
                                   float* __restrict__ agg) {
    constexpr int CH = DIM / 4;                 // float4 chunks per edge
    constexpr int SH = (DIM == 128) ? 5 : 6;    // log2(CH)
    unsigned gid = blockIdx.x * blockDim.x + threadIdx.x;
    if (gid >= (unsigned)NE * CH) return;
    int e = (int)(gid >> SH);
    int c = (int)(gid & (CH - 1));
    int s = src[e], d = dst[e];
    float4 v = *(const float4*)(h + (size_t)s * DIM + 4 * c);
    float* ap = agg + (size_t)d * DIM + 4 * c;
    __hip_atomic_fetch_add(ap + 0, v.x, __ATOMIC_RELAXED, __HIP_MEMORY_SCOPE_AGENT);
    __hip_atomic_fetch_add(ap + 1, v.y, __ATOMIC_RELAXED, __HIP_MEMORY_SCOPE_AGENT);
    __hip_atomic_fetch_add(ap + 2, v.z, __ATOMIC_RELAXED, __HIP_MEMORY_SCOPE_AGENT);
    __hip_atomic_fetch_add(ap + 3, v.w, __ATOMIC_RELAXED, __HIP_MEMORY_SCOPE_AGENT);
}

// ---------------- fused (h+agg) @ w1 -> leaky -> @ w2 -> BN (+leaky) ----------------
// Block: 256 threads (8 waves), owns 64 node rows.
// Wave tiling: 2 M-groups (32 rows) x 4 N-groups (64 cols); each wave holds
// 2 A-frags x 4 B-frags -> 8 WMMA per k-step, each B-frag reused twice.
template <int K1>
__global__ __launch_bounds__(256, 1) void gin_mlp_kernel(
    const float* __restrict__ h, const float* __restrict__ agg,
    const v2f* __restrict__ w1p, const float* __restrict__ b1,
    const v2f* __restrict__ w2p, const float* __restrict__ b2,
    const float* __restrict__ gm, const float* __restrict__ be,
    const float* __restrict__ mu, const float* __restrict__ va,
    float* __restrict__ out, int last)
{
    __shared__ float S[64 * LDSTR];   // A-tile for GEMM1, reused for intermediate T

    const int tid  = threadIdx.x;
    const int lane = tid & 31;
    const int wave = tid >> 5;
    const int l15  = lane & 15;
    const int lhi  = lane >> 4;        // K-half / M-half selector per ISA frag layout
    const int mg   = wave >> 2;        // 0..1 -> 32-row group
    const int n0   = (wave & 3) * 64;  // column group base
    const int row0 = blockIdx.x * 64;

    // ---- stage A = h + agg (64 x K1), zero-padded past N_NODES ----
    {
        constexpr int CH = K1 / 4;
        for (int idx = tid; idx < 64 * CH; idx += 256) {
            int r = idx / CH;
            int c = idx - r * CH;
            float4 val = make_float4(0.f, 0.f, 0.f, 0.f);
            int grow = row0 + r;
            if (grow < NN) {
                float4 hv = *(const float4*)(h   + (size_t)grow * K1 + 4 * c);
                float4 av = *(const float4*)(agg + (size_t)grow * K1 + 4 * c);
                val.x = hv.x + av.x; val.y = hv.y + av.y;
                val.z = hv.z + av.z; val.w = hv.w + av.w;
            }
            *(float4*)(&S[r * LDSTR + 4 * c]) = val;
        }
    }
    __syncthreads();

    const int mr0 = 32 * mg + l15;     // A-frag rows for this lane
    const int mr1 = mr0 + 16;

    // ---- GEMM1: T = A @ w1 ----
    v8f acc[2][4];
    #pragma unroll
    for (int i = 0; i < 2; ++i)
        #pragma unroll
        for (int t = 0; t < 4; ++t)
            acc[i][t] = (v8f){0.f,0.f,0.f,0.f,0.f,0.f,0.f,0.f};

    for (int k = 0; k < K1; k += 4) {
        int kb = k + 2 * lhi;
        v2f a0 = *(const v2f*)(&S[mr0 * LDSTR + kb]);
        v2f a1 = *(const v2f*)(&S[mr1 * LDSTR + kb]);
        int krow = (k >> 1) + lhi;     // packed-pair row index
        #pragma unroll
        for (int t = 0; t < 4; ++t) {
            int col = n0 + t * 16 + l15;
            v2f b = *(const v2f*)(&w1p[(size_t)krow * HID + col]);
            acc[0][t] = __builtin_amdgcn_wmma_f32_16x16x4_f32(
                false, a0, false, b, (short)0, acc[0][t], false, false);
            acc[1][t] = __builtin_amdgcn_wmma_f32_16x16x4_f32(
                false, a1, false, b, (short)0, acc[1][t], false, false);
        }
    }
    __syncthreads();   // everyone done reading A from S

    // ---- bias + leaky, write T back into S ----
    #pragma unroll
    for (int i = 0; i < 2; ++i) {
        #pragma unroll
        for (int t = 0; t < 4; ++t) {
            int col = n0 + t * 16 + l15;
            float bias = b1[col];
            #pragma unroll
            for (int r = 0; r < 8; ++r) {
                float z = acc[i][t][r] + bias;
                z = (z > 0.f) ? z : 0.01f * z;
                int rr = 32 * mg + 16 * i + r + 8 * lhi;  // C/D layout: VGPR r -> rows r, r+8
                S[rr * LDSTR + col] = z;
            }
        }
    }
    __syncthreads();

    // ---- GEMM2: Z = T @ w2 ----
    #pragma unroll
    for (int i = 0; i < 2; ++i)
        #pragma unroll
        for (int t = 0; t < 4; ++t)
            acc[i][t] = (v8f){0.f,0.f,0.f,0.f,0.f,0.f,0.f,0.f};

    for (int k = 0; k < HID; k += 4) {
        int kb = k + 2 * lhi;
        v2f a0 = *(const v2f*)(&S[mr0 * LDSTR + kb]);
        v2f a1 = *(const v2f*)(&S[mr1 * LDSTR + kb]);
        int krow = (k >> 1) + lhi;
        #pragma unroll
        for (int t = 0; t < 4; ++t) {
            int col = n0 + t * 16 + l15;
            v2f b = *(const v2f*)(&w2p[(size_t)krow * HID + col]);
            acc[0][t] = __builtin_amdgcn_wmma_f32_16x16x4_f32(
                false, a0, false, b, (short)0, acc[0][t], false, false);
            acc[1][t] = __builtin_amdgcn_wmma_f32_16x16x4_f32(
                false, a1, false, b, (short)0, acc[1][t], false, false);
        }
    }

    // ---- bias + BN (+leaky) + store ----
    #pragma unroll
    for (int i = 0; i < 2; ++i) {
        #pragma unroll
        for (int t = 0; t < 4; ++t) {
            int col     = n0 + t * 16 + l15;
            float bias  = b2[col];
            float scale = gm[col] * rsqrtf(va[col] + 1e-5f);
            float shift = be[col] - mu[col] * scale;
            #pragma unroll
            for (int r = 0; r < 8; ++r) {
                float z = (acc[i][t][r] + bias) * scale + shift;
                if (!last) z = (z > 0.f) ? z : 0.01f * z;
                int grow = row0 + 32 * mg + 16 * i + r + 8 * lhi;
                if (grow < NN) out[(size_t)grow * HID + col] = z;
            }
        }
    }
}

extern "C" void kernel_launch(void* const* d_in, const int* in_sizes, int n_in,
                              void* d_out, int out_size, void* d_ws, size_t ws_size,
                              hipStream_t stream) {
    const float* x    = (const float*)d_in[0];
    const int*   src  = (const int*)  d_in[1];
    const int*   dst  = (const int*)  d_in[2];
    const float* w1_0 = (const float*)d_in[3];
    const float* b1_0 = (const float*)d_in[4];
    const float* w2_0 = (const float*)d_in[5];
    const float* b2_0 = (const float*)d_in[6];
    const float* g0   = (const float*)d_in[7];
    const float* be0  = (const float*)d_in[8];
    const float* m0   = (const float*)d_in[9];
    const float* v0   = (const float*)d_in[10];
    const float* w1_r = (const float*)d_in[11];
    const float* b1_r = (const float*)d_in[12];
    const float* w2_r = (const float*)d_in[13];
    const float* b2_r = (const float*)d_in[14];
    const float* g_r  = (const float*)d_in[15];
    const float* be_r = (const float*)d_in[16];
    const float* m_r  = (const float*)d_in[17];
    const float* v_r  = (const float*)d_in[18];

    float* out = (float*)d_out;
    float* B0  = (float*)d_ws;                 // NN*HID floats (51.2 MB)

    // packed weights live after B0 in ws (~1.44 MB total)
    float2* w1_0p = (float2*)(B0 + (size_t)NN * HID);   //  64*256 pairs
    float2* w2_0p = w1_0p +  64 * HID;                  // 128*256 pairs
    float2* w1_rp = w2_0p + 128 * HID;                  // 2 layers * 128*256 pairs
    float2* w2_rp = w1_rp + 256 * HID;                  // 2 layers * 128*256 pairs

    const dim3 blk(256);
    const int mlpBlocks = (NN + 63) / 64;

    // ---- pack all weights (k-pair interleave) ----
    gin_pack_kernel<<<( 64 * HID + 255) / 256, blk, 0, stream>>>(w1_0, w1_0p,  64);
    gin_pack_kernel<<<(128 * HID + 255) / 256, blk, 0, stream>>>(w2_0, w2_0p, 128);
    gin_pack_kernel<<<(256 * HID + 255) / 256, blk, 0, stream>>>(w1_r, w1_rp, 256); // 2x(256x256)
    gin_pack_kernel<<<(256 * HID + 255) / 256, blk, 0, stream>>>(w2_r, w2_rp, 256);

    // ---- layer 0: agg->B0 (128-wide), hA -> d_out ----
    {
        int n4 = NN * 128 / 4;
        gin_zero_kernel<<<(n4 + 255) / 256, blk, 0, stream>>>((float4*)B0, n4);
        unsigned work = (unsigned)NE * 32;
        gin_scatter_kernel<128><<<(work + 255) / 256, blk, 0, stream>>>(x, src, dst, B0);
        gin_mlp_kernel<128><<<mlpBlocks, blk, 0, stream>>>(
            x, B0, (const v2f*)w1_0p, b1_0, (const v2f*)w2_0p, b2_0,
            g0, be0, m0, v0, out, 0);
    }
    // ---- layer 1: agg->B0 (256-wide), hB -> B0 (out aliases agg: per-block row ownership) ----
    {
        int n4 = NN * HID / 4;
        gin_zero_kernel<<<(n4 + 255) / 256, blk, 0, stream>>>((float4*)B0, n4);
        unsigned work = (unsigned)NE * 64;
        gin_scatter_kernel<256><<<(work + 255) / 256, blk, 0, stream>>>(out, src, dst, B0);
        gin_mlp_kernel<256><<<mlpBlocks, blk, 0, stream>>>(
            out, B0, (const v2f*)w1_rp, b1_r, (const v2f*)w2_rp, b2_r,
            g_r, be_r, m_r, v_r, B0, 0);
    }
    // ---- layer 2: hA in d_out is dead -> agg->d_out, final -> d_out ----
    {
        int n4 = NN * HID / 4;
        gin_zero_kernel<<<(n4 + 255) / 256, blk, 0, stream>>>((float4*)out, n4);
        unsigned work = (unsigned)NE * 64;
        gin_scatter_kernel<256><<<(work + 255) / 256, blk, 0, stream>>>(B0, src, dst, out);
        gin_mlp_kernel<256><<<mlpBlocks, blk, 0, stream>>>(
            B0, out, (const v2f*)(w1_rp + 128 * HID), b1_r + 256,
            (const v2f*)(w2_rp + 128 * HID), b2_r + 256,
            g_r + 256, be_r + 256, m_r + 256, v_r + 256, out, 1);
    }
}